// ATSSPostProcessor_56676388438749
// MI455X (gfx1250) — compile-verified
//
#include <hip/hip_runtime.h>
#include <math.h>

// ---------------------------------------------------------------------------
// ATSS post-processor for MI455X (gfx1250).
// Memory-bound scan (104MB logits, ~4.5us/pass @ 23.3TB/s, 2nd pass L2-resident
// in the 192MB L2), latency-bound NMS tail (1 workgroup/image, entire working
// set in CDNA5's 320KB LDS). No matmul content -> no WMMA by design; CDNA5
// paths used: async global<->LDS copies (ASYNCcnt + s_wait_asynccnt), wave32
// sizing, LDS atomics, global prefetch.
// ---------------------------------------------------------------------------

#define NIMG 16
#define NLVL 5
#define NCLS 80
#define TOPN 1000
#define MCAND (NLVL * TOPN)          // 5000 candidates per image
#define NBINS 1024
#define OUTN 100
#define NMS_TH 0.6f
#define CLS_OFF 2000.0f
#define SCALE_CLAMP 4.135166556742356f     // log(1000/16)
#define LOGIT_T (-2.9444389791664403f)     // logit(0.05)
#define HIST_HI 12.0f
#define HIST_INVW ((float)NBINS / (HIST_HI - (LOGIT_T)))

// ---- workspace layout (bytes; every sub-buffer & per-image stride 16B) ----
#define WS_HIST   0                                     // u32[16*5*1024]
#define WS_CUT    (WS_HIST + NIMG*NLVL*NBINS*4)         // int[16*5*4]: cutbin, base, cntA, cntB
#define WS_SCORE  (WS_CUT + NIMG*NLVL*4*4)              // f32[16*5000]
#define WS_BOX    (WS_SCORE + NIMG*MCAND*4)             // float4[16*5000]
#define WS_CLS    (WS_BOX + NIMG*MCAND*16)              // int[16*5000]
#define WS_SIDX   (WS_CLS + NIMG*MCAND*4)               // int[16*5000]
#define WS_ZERO_WORDS (WS_SIDX / 4)                     // zero hist..candCls

// ---------------------------------------------------------------------------
__global__ void k_init(uint4* __restrict__ ws4, int n4) {
  int i = blockIdx.x * blockDim.x + threadIdx.x;
  int stride = gridDim.x * blockDim.x;
  uint4 z; z.x = 0u; z.y = 0u; z.z = 0u; z.w = 0u;
  for (; i < n4; i += stride) ws4[i] = z;
}

// Per (image, level) logit histogram. Streaming float4 reads, LDS atomics.
__global__ void k_hist(const float* __restrict__ cls, unsigned* __restrict__ hist,
                       int lvl, int flat) {
  __shared__ unsigned lh[NBINS];
  const int tid = threadIdx.x;
  const int img = blockIdx.y;
  for (int b = tid; b < NBINS; b += blockDim.x) lh[b] = 0u;
  __syncthreads();
  const float4* p = (const float4*)(cls + (size_t)img * flat);
  const int n4 = flat >> 2;
  int i = blockIdx.x * blockDim.x + tid;
  const int stride = gridDim.x * blockDim.x;
  for (; i < n4; i += stride) {
    __builtin_prefetch(p + i + stride, 0, 0);   // -> global_prefetch_b8
    float4 v = p[i];
    float xs[4] = {v.x, v.y, v.z, v.w};
#pragma unroll
    for (int q = 0; q < 4; ++q) {
      float x = xs[q];
      if (x > LOGIT_T) {
        int b = (int)((x - LOGIT_T) * HIST_INVW);
        b = (b > NBINS - 1) ? (NBINS - 1) : b;
        atomicAdd(&lh[b], 1u);
      }
    }
  }
  __syncthreads();
  unsigned* gh = hist + ((size_t)img * NLVL + lvl) * NBINS;
  for (int b = tid; b < NBINS; b += blockDim.x) {
    unsigned c = lh[b];
    if (c) atomicAdd(&gh[b], c);
  }
}

// Suffix-scan the histogram; find cutoff bin so that count above >= TOPN.
__global__ void k_cutoff(const unsigned* __restrict__ hist, int* __restrict__ cut) {
  __shared__ unsigned s[NBINS];
  const int row = blockIdx.x;          // img*NLVL + lvl
  const int t = threadIdx.x;           // blockDim == NBINS
  s[t] = hist[(size_t)row * NBINS + t];
  __syncthreads();
  for (int off = 1; off < NBINS; off <<= 1) {
    unsigned v = (t + off < NBINS) ? s[t + off] : 0u;
    __syncthreads();
    s[t] += v;
    __syncthreads();
  }
  // s[t] = count of survivors in bins [t, NBINS)
  unsigned st = s[t];
  unsigned snext = (t + 1 < NBINS) ? s[t + 1] : 0u;
  int* c = cut + row * 4;
  if (st >= (unsigned)TOPN && snext < (unsigned)TOPN) { c[0] = t; c[1] = (int)snext; }
  if (t == 0 && s[0] < (unsigned)TOPN) { c[0] = 0; c[1] = (int)s[1]; }
}

// Second pass: select survivors, assign slots, decode+clip boxes, score.
__global__ void k_compact(const float* __restrict__ cls, const float* __restrict__ reg,
                          const float* __restrict__ anc, const int* __restrict__ imsz,
                          int* __restrict__ cut, float* __restrict__ candScore,
                          float4* __restrict__ candBox, int* __restrict__ candCls,
                          int lvl, int HW) {
  const int img = blockIdx.y;
  const int flat = HW * NCLS;
  int* crow = cut + (img * NLVL + lvl) * 4;
  const int cutbin = crow[0];
  const int base = crow[1];
  const float hImg = (float)imsz[img * 2 + 0];
  const float wImg = (float)imsz[img * 2 + 1];
  const float* clsImg = cls + (size_t)img * flat;
  const float* regImg = reg + (size_t)img * 4 * HW;
  const float4* anc4 = (const float4*)anc;
  int i = blockIdx.x * blockDim.x + threadIdx.x;
  const int stride = gridDim.x * blockDim.x;
  for (; i < flat; i += stride) {
    float x = clsImg[i];
    if (!(x > LOGIT_T)) continue;
    int b = (int)((x - LOGIT_T) * HIST_INVW);
    b = (b > NBINS - 1) ? (NBINS - 1) : b;
    if (b < cutbin) continue;
    int pos;
    if (b > cutbin) {
      pos = atomicAdd(&crow[2], 1);          // strictly-above: slots [0, base)
    } else {
      int j = atomicAdd(&crow[3], 1);        // boundary bin: slots [base, TOPN)
      if (j >= TOPN - base) continue;
      pos = base + j;
    }
    if (pos >= TOPN) continue;  // safety
    int c = i / HW;
    int hw = i - c * HW;
    float4 a = anc4[hw];
    float aw = a.z - a.x + 1.0f, ah = a.w - a.y + 1.0f;
    float cx = a.x + 0.5f * aw, cy = a.y + 0.5f * ah;
    float dx = regImg[hw] * 0.1f;
    float dy = regImg[HW + hw] * 0.1f;
    float dw = fminf(regImg[2 * HW + hw] * 0.2f, SCALE_CLAMP);
    float dh = fminf(regImg[3 * HW + hw] * 0.2f, SCALE_CLAMP);
    float pcx = dx * aw + cx, pcy = dy * ah + cy;
    float pw = __expf(dw) * aw, ph = __expf(dh) * ah;
    float x1 = pcx - 0.5f * (pw - 1.0f);
    float y1 = pcy - 0.5f * (ph - 1.0f);
    float x2 = pcx + 0.5f * (pw - 1.0f);
    float y2 = pcy + 0.5f * (ph - 1.0f);
    x1 = fminf(fmaxf(x1, 0.0f), wImg);
    x2 = fminf(fmaxf(x2, 0.0f), wImg);
    y1 = fminf(fmaxf(y1, 0.0f), hImg);
    y2 = fminf(fmaxf(y2, 0.0f), hImg);
    float sig = 1.0f / (1.0f + __expf(-x));
    float sc = sqrtf(sig);
    if (!((x2 - x1) > 0.0f && (y2 - y1) > 0.0f)) sc = 0.0f;
    size_t g = (size_t)img * MCAND + (size_t)lvl * TOPN + pos;
    candScore[g] = sc;
    candCls[g] = c;
    candBox[g] = make_float4(x1, y1, x2, y2);
  }
}

// Per-image bitonic sort (descending) of 5000 scores padded to 8192 in LDS.
// Keys streamed in with async global->LDS b128 copies; sorted indices streamed
// out with async LDS->global b128 stores (both ASYNCcnt-tracked).
__global__ void k_sort(const float* __restrict__ candScore, int* __restrict__ sidx) {
  extern __shared__ char smem[];
  float* key = (float*)smem;              // 8192 * 4B
  int* val = (int*)(smem + 8192 * 4);     // 8192 * 4B
  const int img = blockIdx.x, tid = threadIdx.x;
  const float* sc = candScore + (size_t)img * MCAND;

  // Async bulk copy of 5000 scores (1250 x b128) into LDS key[0..4999].
  for (int t = tid; t < MCAND / 4; t += blockDim.x) {
    unsigned lds = (unsigned)(unsigned long long)(uintptr_t)(&key[t * 4]);
    unsigned long long ga = (unsigned long long)(uintptr_t)(sc + t * 4);
    asm volatile("global_load_async_to_lds_b128 %0, %1, off"
                 :: "v"(lds), "v"(ga) : "memory");
  }
#pragma unroll
  for (int e = 0; e < 8; ++e) {
    int i = tid + e * 1024;
    if (i >= MCAND) key[i] = -1.0f;   // sentinel pads (sort descending)
    val[i] = i;
  }
  asm volatile("s_wait_asynccnt 0x0" ::: "memory");
  __syncthreads();

  for (int k = 2; k <= 8192; k <<= 1) {
    for (int j = k >> 1; j > 0; j >>= 1) {
#pragma unroll
      for (int e = 0; e < 8; ++e) {
        int i = tid + e * 1024;
        int ixj = i ^ j;
        if (ixj > i) {
          bool up = ((i & k) == 0);
          float a = key[i], b = key[ixj];
          bool sw = up ? (a < b) : (a > b);   // descending overall
          if (sw) {
            key[i] = b; key[ixj] = a;
            int t = val[i]; val[i] = val[ixj]; val[ixj] = t;
          }
        }
      }
      __syncthreads();
    }
  }

  // Async bulk store of sorted indices val[0..4999] (1250 x b128) to global.
  int* out = sidx + (size_t)img * MCAND;
  for (int t = tid; t < MCAND / 4; t += blockDim.x) {
    unsigned lds = (unsigned)(unsigned long long)(uintptr_t)(&val[t * 4]);
    unsigned long long ga = (unsigned long long)(uintptr_t)(out + t * 4);
    asm volatile("global_store_async_from_lds_b128 %0, %1, off"
                 :: "v"(ga), "v"(lds) : "memory");
  }
  asm volatile("s_wait_asynccnt 0x0" ::: "memory");
}

// Greedy class-aware NMS per image; boxes staged into LDS with CDNA5
// async global->LDS copies (ASYNCcnt), then O(M^2) suppression, emit top-100.
__global__ void k_nms(const float4* __restrict__ candBox, const float* __restrict__ candScore,
                      const int* __restrict__ candCls, const int* __restrict__ sidx,
                      float* __restrict__ outF, int* __restrict__ outC) {
  extern __shared__ char smem[];
  float4* box4 = (float4*)smem;                          // 5000*16 = 80000B
  float* area = (float*)(smem + 80000);                  // 20000B
  float* ssc = (float*)(smem + 100000);                  // 20000B
  int* scls = (int*)(smem + 120000);                     // 20000B
  unsigned char* sup = (unsigned char*)(smem + 140000);  // 5000B
  const int tid = threadIdx.x;
  const int img = blockIdx.x;
  const int* sIdx = sidx + (size_t)img * MCAND;

  // Phase 1: async gather of sorted boxes into LDS (b128, ASYNCcnt-tracked).
  for (int r = tid; r < MCAND; r += blockDim.x) {
    int idx = sIdx[r];
    unsigned lds = (unsigned)(unsigned long long)(uintptr_t)(&box4[r]);
    unsigned long long ga =
        (unsigned long long)(uintptr_t)(candBox + (size_t)img * MCAND + idx);
    asm volatile("global_load_async_to_lds_b128 %0, %1, off"
                 :: "v"(lds), "v"(ga) : "memory");
  }
  asm volatile("s_wait_asynccnt 0x0" ::: "memory");
  __syncthreads();

  // Phase 2: class offsets, areas, scores, initial suppression flags.
  for (int r = tid; r < MCAND; r += blockDim.x) {
    int idx = sIdx[r];
    float4 b = box4[r];
    int c = candCls[(size_t)img * MCAND + idx];
    float off = (float)c * CLS_OFF;
    b.x += off; b.y += off; b.z += off; b.w += off;
    box4[r] = b;
    area[r] = (b.z - b.x) * (b.w - b.y);
    float s = candScore[(size_t)img * MCAND + idx];
    ssc[r] = s;
    scls[r] = c;
    sup[r] = (s > 0.0f) ? 0 : 1;
  }
  __syncthreads();

  // Phase 3: greedy NMS (serial over i, parallel over j).
  for (int i = 0; i < MCAND; ++i) {
    __syncthreads();
    if (sup[i]) continue;
    float4 bi = box4[i];
    float ai = area[i];
    for (int j = i + 1 + tid; j < MCAND; j += blockDim.x) {
      if (sup[j]) continue;
      float4 bj = box4[j];
      float xx1 = fmaxf(bi.x, bj.x), yy1 = fmaxf(bi.y, bj.y);
      float xx2 = fminf(bi.z, bj.z), yy2 = fminf(bi.w, bj.w);
      float inter = fmaxf(xx2 - xx1, 0.0f) * fmaxf(yy2 - yy1, 0.0f);
      float iou = inter / (ai + area[j] - inter + 1e-9f);
      if (iou > NMS_TH) sup[j] = 1;
    }
  }
  __syncthreads();

  // Phase 4: first 100 kept (sorted order == descending score) -> output.
  if (tid == 0) {
    float* ob = outF + (size_t)img * OUTN * 4;
    float* os = outF + (size_t)NIMG * OUTN * 4 + (size_t)img * OUTN;
    int* oc = outC + (size_t)img * OUTN;
    int k = 0;
    for (int r = 0; r < MCAND && k < OUTN; ++r) {
      if (!sup[r]) {
        float off = (float)scls[r] * CLS_OFF;
        float4 b = box4[r];
        ob[k * 4 + 0] = b.x - off;
        ob[k * 4 + 1] = b.y - off;
        ob[k * 4 + 2] = b.z - off;
        ob[k * 4 + 3] = b.w - off;
        os[k] = ssc[r];
        oc[k] = scls[r];
        ++k;
      }
    }
    for (; k < OUTN; ++k) {
      ob[k * 4 + 0] = 0.0f; ob[k * 4 + 1] = 0.0f;
      ob[k * 4 + 2] = 0.0f; ob[k * 4 + 3] = 0.0f;
      os[k] = 0.0f; oc[k] = 0;
    }
  }
}

// ---------------------------------------------------------------------------
extern "C" void kernel_launch(void* const* d_in, const int* in_sizes, int n_in,
                              void* d_out, int out_size, void* d_ws, size_t ws_size,
                              hipStream_t stream) {
  (void)in_sizes; (void)n_in; (void)out_size; (void)ws_size;
  static const int HWs[NLVL] = {15200, 3800, 950, 247, 70};

  char* ws = (char*)d_ws;
  unsigned* hist = (unsigned*)(ws + WS_HIST);
  int* cut = (int*)(ws + WS_CUT);
  float* candScore = (float*)(ws + WS_SCORE);
  float4* candBox = (float4*)(ws + WS_BOX);
  int* candCls = (int*)(ws + WS_CLS);
  int* sidx = (int*)(ws + WS_SIDX);
  const int* imsz = (const int*)d_in[15];     // image_sizes [16,2] int32
  float* outF = (float*)d_out;
  int* outC = (int*)(outF + NIMG * OUTN * 4 + NIMG * OUTN);

  k_init<<<dim3(512), dim3(256), 0, stream>>>((uint4*)ws, WS_ZERO_WORDS / 4);

  for (int l = 0; l < NLVL; ++l) {
    const float* clsP = (const float*)d_in[l * 3 + 0];
    int flat = HWs[l] * NCLS;
    int n4 = flat / 4;
    int bl = (n4 + 255) / 256; if (bl > 128) bl = 128;
    k_hist<<<dim3(bl, NIMG), dim3(256), 0, stream>>>(clsP, hist, l, flat);
  }

  k_cutoff<<<dim3(NIMG * NLVL), dim3(NBINS), 0, stream>>>(hist, cut);

  for (int l = 0; l < NLVL; ++l) {
    const float* clsP = (const float*)d_in[l * 3 + 0];
    const float* regP = (const float*)d_in[l * 3 + 1];
    const float* ancP = (const float*)d_in[l * 3 + 2];
    int flat = HWs[l] * NCLS;
    int bl = (flat + 255) / 256; if (bl > 256) bl = 256;
    k_compact<<<dim3(bl, NIMG), dim3(256), 0, stream>>>(
        clsP, regP, ancP, imsz, cut, candScore, candBox, candCls, l, HWs[l]);
  }

  k_sort<<<dim3(NIMG), dim3(1024), 8192 * 8, stream>>>(candScore, sidx);
  k_nms<<<dim3(NIMG), dim3(1024), 145024, stream>>>(candBox, candScore, candCls,
                                                    sidx, outF, outC);
}